// AttentionResidual_68865505624122
// MI455X (gfx1250) — compile-verified
//
#include <hip/hip_runtime.h>

// Problem shape (fixed by setup_inputs):
#define MAX_T 16
#define BB    8
#define SSZ   2048
#define DD    1024
#define BSD   (BB * SSZ * DD)     // 16,777,216 floats: t-stride of entries
#define NTHREADS 256

typedef float v2f __attribute__((ext_vector_type(2)));
typedef float v8f __attribute__((ext_vector_type(8)));

// Async-to/from-LDS builtins take typed int4 pointers:
//   arg0: int __vector(4) __device__* (addrspace(1))
//   arg1: int __vector(4) __shared__* (addrspace(3))
typedef int v4i __attribute__((vector_size(16)));
typedef __attribute__((address_space(1))) v4i v4i_g;
typedef __attribute__((address_space(3))) v4i v4i_l;

// ---- LDS layout (float indices) ----
#define E_OFF   0            // 16 x 1024 tile                 (16384 floats)
#define QS_OFF  16384        // qs = proj[row]*norm_scale      ( 1024 floats)
#define PSS_OFF 17408        // partial sum-of-squares 16x16   (  256 floats)
#define PQD_OFF 17664        // partial q.e            16x16   (  256 floats)
#define LOG_OFF 17920        // logits                         (   16 floats)
#define W_OFF   17936        // softmax weights                (   16 floats)
#define OUT_OFF 17952        // output staging (16B aligned)   ( 1024 floats)
#define LDS_FLOATS (OUT_OFF + DD)
#define LDS_BYTES  (LDS_FLOATS * 4)

#if __has_builtin(__builtin_amdgcn_global_load_async_to_lds_b128)
#define HAVE_ASYNC_LD 1
#endif
#if __has_builtin(__builtin_amdgcn_global_store_async_from_lds_b128)
#define HAVE_ASYNC_ST 1
#endif
#if __has_builtin(__builtin_amdgcn_wmma_f32_16x16x4_f32)
#define HAVE_WMMA_F32 1
#endif

static __device__ __forceinline__ void wait_async0() {
#if __has_builtin(__builtin_amdgcn_s_wait_asynccnt)
  __builtin_amdgcn_s_wait_asynccnt(0);
#else
  asm volatile("s_wait_asynccnt 0" ::: "memory");
#endif
}

__global__ __launch_bounds__(NTHREADS) void
attn_depth_kernel(const float* __restrict__ entries,
                  const float* __restrict__ proj,
                  const float* __restrict__ nscale,
                  const int*   __restrict__ p_nact,
                  const int*   __restrict__ p_bidx,
                  float*       __restrict__ out) {
  extern __shared__ float smem[];

  const int tid  = threadIdx.x;
  const int lane = tid & 31;
  const int wv   = tid >> 5;                // 8 waves
  const size_t bs   = blockIdx.x;           // b*S + s, 0..16383
  const size_t base = bs * (size_t)DD;

  const int nact = *p_nact;
  int row = *p_bidx;
  if (row > MAX_T - 1) row = MAX_T - 1;
  if (row < 0) row = 0;

  // ---------- Stage 1: DMA the 16x1024 fp32 tile into LDS ----------
#if defined(HAVE_ASYNC_LD)
  {
    // each wave moves 2 rows; 128-bit per lane, 8 issues per row
    #pragma unroll
    for (int r = 0; r < 2; ++r) {
      const int t = wv * 2 + r;
      const float* g = entries + (size_t)t * (size_t)BSD + base;
      float* l = &smem[E_OFF + t * DD];
      #pragma unroll
      for (int j = 0; j < 8; ++j) {
        const int f = (lane + j * 32) * 4;   // float offset, 16B granular
        __builtin_amdgcn_global_load_async_to_lds_b128(
            (v4i_g*)(void*)(g + f), (v4i_l*)(void*)(l + f), 0, 0);
      }
    }
  }
#else
  for (int i = tid; i < MAX_T * DD / 4; i += NTHREADS) {
    const int t = i >> 8;                    // 256 float4 per row
    const int f = (i & 255) * 4;
    const float4 v = *(const float4*)(entries + (size_t)t * (size_t)BSD + base + f);
    *(float4*)&smem[E_OFF + t * DD + f] = v;
  }
#endif

  // qs = proj[row] * norm_scale (tiny, L2-resident across all blocks)
  for (int d = tid; d < DD; d += NTHREADS)
    smem[QS_OFF + d] = proj[row * DD + d] * nscale[d];

#if defined(HAVE_ASYNC_LD)
  wait_async0();
#endif
  __syncthreads();

  // ---------- Stage 2: per-t sum(e^2) and sum(qs*e) partials ----------
  {
    const int t2 = tid >> 4;                 // 16 threads per t-row
    const int g  = tid & 15;
    float ss = 0.f, qd = 0.f;
    #pragma unroll 8
    for (int i = 0; i < 64; ++i) {
      const int d = g + (i << 4);
      const float e = smem[E_OFF + t2 * DD + d];
      ss = fmaf(e, e, ss);
      qd = fmaf(smem[QS_OFF + d], e, qd);
    }
    smem[PSS_OFF + t2 * 16 + g] = ss;
    smem[PQD_OFF + t2 * 16 + g] = qd;
  }
  __syncthreads();

  // ---------- Stage 3: logits (deterministic reduction) ----------
  if (tid < MAX_T) {
    float s2 = 0.f, q2 = 0.f;
    #pragma unroll
    for (int i = 0; i < 16; ++i) {
      s2 += smem[PSS_OFF + tid * 16 + i];
      q2 += smem[PQD_OFF + tid * 16 + i];
    }
    // logit = dot(qs, e) / sqrt(mean(e^2) + eps)
    float lg = q2 * rsqrtf(s2 * (1.0f / (float)DD) + 1e-5f);
    if (tid >= nact) lg = -1e9f;
    smem[LOG_OFF + tid] = lg;
  }
  __syncthreads();

  // ---------- softmax over the 16 depths -> weights in LDS ----------
  if (tid < MAX_T) {
    float m = -3.402823466e+38f;
    float sum = 0.f;
    #pragma unroll
    for (int i = 0; i < MAX_T; ++i) m = fmaxf(m, smem[LOG_OFF + i]);
    #pragma unroll
    for (int i = 0; i < MAX_T; ++i) sum += __expf(smem[LOG_OFF + i] - m);
    // dynamic LDS address (not a register-file gather) for own logit
    smem[W_OFF + tid] = __expf(smem[LOG_OFF + tid] - m) / sum;
  }
  __syncthreads();

  // ---------- Stage 4: out[d] = sum_t w[t]*E[t][d] via fp32 WMMA ----------
#if defined(HAVE_WMMA_F32)
  {
    // D(16x16) = A(16x4) x B(4x16) + C, K accumulated in 4 chunks of 4.
    // A rows all equal the weight vector chunk => every row of D is the result.
    // A layout: lanes 0-15 hold K=0,1 ; lanes 16-31 hold K=2,3.
    const int n  = lane & 15;
    const int kh = lane >> 4;

    // Hoist A operands: even-aligned weight pairs from LDS (2 distinct
    // addresses per wave -> broadcast-friendly, no register indexing).
    v2f aop[4];
    #pragma unroll
    for (int c = 0; c < 4; ++c) {
      aop[c][0] = smem[W_OFF + 4 * c + 2 * kh];
      aop[c][1] = smem[W_OFF + 4 * c + 2 * kh + 1];
    }

    #pragma unroll
    for (int j = 0; j < 8; ++j) {
      const int d0 = wv * 128 + j * 16;
      v8f acc = {0.f, 0.f, 0.f, 0.f, 0.f, 0.f, 0.f, 0.f};
      #pragma unroll
      for (int c = 0; c < 4; ++c) {
        v2f b;
        b[0] = smem[E_OFF + (4 * c + 2 * kh    ) * DD + d0 + n];
        b[1] = smem[E_OFF + (4 * c + 2 * kh + 1) * DD + d0 + n];
        acc = __builtin_amdgcn_wmma_f32_16x16x4_f32(
            false, aop[c], false, b, (short)0, acc, false, false);
      }
      if (lane < 16) smem[OUT_OFF + d0 + n] = acc[0];   // row M=0 of D
    }
  }
#else
  for (int d = tid; d < DD; d += NTHREADS) {
    float s = 0.f;
    #pragma unroll
    for (int t = 0; t < MAX_T; ++t)
      s = fmaf(smem[W_OFF + t], smem[E_OFF + t * DD + d], s);
    smem[OUT_OFF + d] = s;
  }
#endif
  __syncthreads();

  // ---------- Stage 5: store 4KB result (one 512B async store per wave) ----------
#if defined(HAVE_ASYNC_ST)
  {
    float* gw = out + base + wv * 128;
    float* lr = &smem[OUT_OFF + wv * 128];
    __builtin_amdgcn_global_store_async_from_lds_b128(
        (v4i_g*)(void*)(gw + lane * 4), (v4i_l*)(void*)(lr + lane * 4), 0, 0);
    wait_async0();
  }
#else
  {
    const float4 v = *(const float4*)&smem[OUT_OFF + wv * 128 + lane * 4];
    *(float4*)(out + base + wv * 128 + lane * 4) = v;
  }
#endif
}

extern "C" void kernel_launch(void* const* d_in, const int* in_sizes, int n_in,
                              void* d_out, int out_size, void* d_ws, size_t ws_size,
                              hipStream_t stream) {
  (void)in_sizes; (void)n_in; (void)out_size; (void)d_ws; (void)ws_size;
  const float* entries = (const float*)d_in[0];
  const float* proj    = (const float*)d_in[1];
  const float* nscale  = (const float*)d_in[2];
  const int*   nact    = (const int*)d_in[3];
  const int*   bidx    = (const int*)d_in[4];
  float* out = (float*)d_out;

  dim3 grid(BB * SSZ);      // one workgroup per (b, s)
  dim3 block(NTHREADS);     // 8 wave32
  hipLaunchKernelGGL(attn_depth_kernel, grid, block, LDS_BYTES, stream,
                     entries, proj, nscale, nact, bidx, out);
}